// MambaLayer_70205535420852
// MI455X (gfx1250) — compile-verified
//
#include <hip/hip_runtime.h>
#include <stdint.h>

// Mamba layer for MI455X (gfx1250, wave32).
// Pipeline: LayerNorm -> WMMA GEMM (W_in) -> causal dwconv+SiLU ->
//           WMMA GEMM (W_xproj) -> WMMA GEMM (W_dt)+softplus ->
//           selective scan (lane-per-state, shuffle reduce) + SiLU(z) gate ->
//           WMMA GEMM (W_out) + residual.
// GEMMs: V_WMMA_F32_16X16X4_F32, double-buffered LDS staged with
// GLOBAL_LOAD_ASYNC_TO_LDS_B128 (ASYNCcnt) overlapping compute.

#define B_SZ    2
#define L_SZ    2048
#define DMODEL  1024
#define DINNER  2048
#define DSTATE  16
#define DCONV   4
#define DTRANK  64
#define XPROJ_N (DTRANK + 2 * DSTATE)   // 96

typedef __attribute__((ext_vector_type(2))) float v2f;
typedef __attribute__((ext_vector_type(8))) float v8f;

__device__ __forceinline__ float silu_f(float v) { return v / (1.f + __expf(-v)); }

// ---------------------------------------------------------------- LayerNorm
__global__ __launch_bounds__(256) void ln_kernel(const float* __restrict__ x,
                                                 const float* __restrict__ g,
                                                 const float* __restrict__ b,
                                                 float* __restrict__ o)
{
    const int row = blockIdx.x;                       // B*L rows
    const float* xr = x + (size_t)row * DMODEL;
    float s = 0.f, q = 0.f;
    for (int c = threadIdx.x; c < DMODEL; c += 256) { float v = xr[c]; s += v; q += v * v; }
#pragma unroll
    for (int off = 16; off > 0; off >>= 1) { s += __shfl_down(s, off); q += __shfl_down(q, off); }
    __shared__ float ss[8], sq[8], smu, srs;
    const int w = threadIdx.x >> 5, ln = threadIdx.x & 31;
    if (ln == 0) { ss[w] = s; sq[w] = q; }
    __syncthreads();
    if (threadIdx.x == 0) {
        float ts = 0.f, tq = 0.f;
        for (int i = 0; i < 8; ++i) { ts += ss[i]; tq += sq[i]; }
        const float mu  = ts / (float)DMODEL;
        const float var = tq / (float)DMODEL - mu * mu;
        smu = mu; srs = rsqrtf(var + 1e-5f);
    }
    __syncthreads();
    const float mu = smu, rs = srs;
    float* orow = o + (size_t)row * DMODEL;
    for (int c = threadIdx.x; c < DMODEL; c += 256)
        orow[c] = (xr[c] - mu) * rs * g[c] + b[c];
}

// ---------------------------------------------------------------- WMMA GEMM
// C[M,N] = A[M,K] * B[K,N]  (row-major fp32, V_WMMA_F32_16X16X4_F32)
// Block tile 128x128, 8 waves in a 4x2 grid, 32x64 per wave (2x4 WMMA tiles).
// Double-buffered LDS; full tiles staged via global_load_async_to_lds_b128.
// epi: 0 = none, 1 = softplus(v + bias[col]), 2 = v + resid[row*ldr+col]
#define BM 128
#define BN 128
#define BK 16
#define SA_STRIDE 20    // m-major A rows; even, mult-of-4, bank-conflict-free b64 reads
#define SB_STRIDE 136   // k-major B rows; 2*136 mod 64 = 16 -> half-wave windows disjoint

__global__ __launch_bounds__(256) void gemm_wmma_f32(
    const float* __restrict__ A, const float* __restrict__ Bm, float* __restrict__ C,
    int M, int N, int K, int lda, int ldb, int ldc,
    int epi, const float* __restrict__ bias, const float* __restrict__ resid, int ldr)
{
    __shared__ float sA[2][BM][SA_STRIDE];
    __shared__ float sB[2][BK][SB_STRIDE];

    const int m0 = blockIdx.y * BM, n0 = blockIdx.x * BN;
    const int w  = threadIdx.x >> 5;          // wave 0..7
    const int ln = threadIdx.x & 15;          // lane % 16
    const int hi = (threadIdx.x & 31) >> 4;   // upper half-wave -> K+2 split
    const int wm = w >> 1, wn = w & 1;        // 4x2 wave grid
    const bool full = (n0 + BN <= N);         // M,K are tile-multiples for all call sites

    const v8f zero = {0.f, 0.f, 0.f, 0.f, 0.f, 0.f, 0.f, 0.f};
    v8f acc[2][4];
#pragma unroll
    for (int i = 0; i < 2; ++i)
#pragma unroll
        for (int j = 0; j < 4; ++j) acc[i][j] = zero;

    auto copy_tile = [&](int k0c, int bufc) {
        if (full) {
            // B tile: BK x BN = 2048 floats = 512 x b128, 2 per thread
#pragma unroll
            for (int t = 0; t < 2; ++t) {
                const int i4 = (int)threadIdx.x + t * 256;
                const int kr = i4 >> 5, q = i4 & 31;           // 32 vec4 per row
                const unsigned lds = (unsigned)(uintptr_t)&sB[bufc][kr][q * 4];
                const unsigned long long g =
                    (unsigned long long)(uintptr_t)(Bm + (size_t)(k0c + kr) * ldb + n0 + q * 4);
                asm volatile("global_load_async_to_lds_b128 %0, %1, off"
                             :: "v"(lds), "v"(g) : "memory");
            }
            // A tile: BM x BK = 2048 floats = 512 x b128, 2 per thread (m-major)
#pragma unroll
            for (int t = 0; t < 2; ++t) {
                const int i4 = (int)threadIdx.x + t * 256;
                const int mr = i4 >> 2, q = i4 & 3;            // 4 vec4 per row
                const unsigned lds = (unsigned)(uintptr_t)&sA[bufc][mr][q * 4];
                const unsigned long long g =
                    (unsigned long long)(uintptr_t)(A + (size_t)(m0 + mr) * lda + k0c + q * 4);
                asm volatile("global_load_async_to_lds_b128 %0, %1, off"
                             :: "v"(lds), "v"(g) : "memory");
            }
        } else {
            // guarded synchronous staging with zero fill (keeps WMMA operands valid)
            for (int idx = threadIdx.x; idx < BK * BN; idx += 256) {
                const int kr = idx / BN, nc = idx % BN;
                const int gn = n0 + nc;
                sB[bufc][kr][nc] = (gn < N) ? Bm[(size_t)(k0c + kr) * ldb + gn] : 0.f;
            }
            for (int idx = threadIdx.x; idx < BM * BK; idx += 256) {
                const int mr = idx / BK, kc = idx % BK;
                sA[bufc][mr][kc] = A[(size_t)(m0 + mr) * lda + k0c + kc];
            }
        }
    };

    int buf = 0;
    copy_tile(0, 0);
    asm volatile("s_wait_asynccnt 0" ::: "memory");
    __syncthreads();

    for (int k0 = 0; k0 < K; k0 += BK) {
        if (k0 + BK < K) copy_tile(k0 + BK, buf ^ 1);   // prefetch next tile (async)

#pragma unroll
        for (int kk = 0; kk < BK; kk += 4) {
            v2f af[2], bf[4];
#pragma unroll
            for (int i = 0; i < 2; ++i)  // A frag: 2 consecutive K per lane -> ds_load_b64
                af[i] = *(const v2f*)&sA[buf][wm * 32 + i * 16 + ln][kk + 2 * hi];
#pragma unroll
            for (int j = 0; j < 4; ++j) {
                const int n = wn * 64 + j * 16 + ln;
                bf[j].x = sB[buf][kk + 2 * hi][n];
                bf[j].y = sB[buf][kk + 1 + 2 * hi][n];
            }
#pragma unroll
            for (int i = 0; i < 2; ++i)
#pragma unroll
                for (int j = 0; j < 4; ++j)
                    acc[i][j] = __builtin_amdgcn_wmma_f32_16x16x4_f32(
                        false, af[i], false, bf[j], (short)0, acc[i][j], false, false);
        }

        asm volatile("s_wait_asynccnt 0" ::: "memory");
        __syncthreads();
        buf ^= 1;
    }

    // epilogue + store: VGPR r holds rows {r, r+8} for half-waves {lo, hi}
#pragma unroll
    for (int i = 0; i < 2; ++i) {
#pragma unroll
        for (int j = 0; j < 4; ++j) {
            const int col = n0 + wn * 64 + j * 16 + ln;
            if (col >= N) continue;
#pragma unroll
            for (int r = 0; r < 8; ++r) {
                const int row = m0 + wm * 32 + i * 16 + 8 * hi + r;
                if (row >= M) continue;
                float v = acc[i][j][r];
                if (epi == 1) {
                    v += bias[col];
                    v = (v > 20.f) ? v : log1pf(__expf(v));   // softplus
                } else if (epi == 2) {
                    v += resid[(size_t)row * ldr + col];
                }
                C[(size_t)row * ldc + col] = v;
            }
        }
    }
}

// ------------------------------------------------- causal dwconv (4-tap) + SiLU
__global__ __launch_bounds__(256) void conv_silu_kernel(const float* __restrict__ xz,
                                                        const float* __restrict__ wgt,
                                                        const float* __restrict__ bias,
                                                        float* __restrict__ u)
{
    const size_t idx = (size_t)blockIdx.x * 256 + threadIdx.x;   // over B*L*DINNER
    const int d = (int)(idx % DINNER);
    const size_t bl = idx / DINNER;
    const int l = (int)(bl % L_SZ);
    const size_t b = bl / L_SZ;
    const float* col = xz + (b * L_SZ) * (size_t)(2 * DINNER) + d;  // u half of xz
    float acc = bias[d];
#pragma unroll
    for (int j = 0; j < DCONV; ++j) {
        const int ll = l - (DCONV - 1) + j;
        if (ll >= 0) acc += col[(size_t)ll * (2 * DINNER)] * wgt[d * DCONV + j];
    }
    u[idx] = silu_f(acc);
}

// -------------------------------------------- selective scan + SiLU(z) gating
// lane = (state n) + 16*(channel parity); each wave runs 2 channels, the
// einsum over n is 4 shfl_xor reductions inside each 16-lane group.
__global__ __launch_bounds__(256) void scan_kernel(
    const float* __restrict__ u, const float* __restrict__ delta,
    const float* __restrict__ xdbl, const float* __restrict__ A_log,
    const float* __restrict__ Dskip, const float* __restrict__ xz,
    float* __restrict__ y)
{
    const int lane = threadIdx.x & 31;
    const int wave = threadIdx.x >> 5;
    const int n    = lane & 15;
    const int hh   = lane >> 4;
    const int chan = wave * 2 + hh;                 // 0..15 within block
    const int blocksPerBatch = DINNER / 16;         // 128
    const int b = blockIdx.x / blocksPerBatch;
    const int d = (blockIdx.x % blocksPerBatch) * 16 + chan;

    const float a  = -__expf(A_log[d * DSTATE + n]);
    const float Dd = Dskip[d];
    float h = 0.f;

    size_t rowD = ((size_t)b * L_SZ) * DINNER + d;                  // delta/u/y index
    size_t rowX = ((size_t)b * L_SZ) * XPROJ_N;                     // x_dbl row
    size_t rowZ = ((size_t)b * L_SZ) * (2 * DINNER) + DINNER + d;   // z half of xz

    for (int l = 0; l < L_SZ; ++l) {
        const float dl = delta[rowD];
        const float ut = u[rowD];
        const float Bt = xdbl[rowX + DTRANK + n];
        const float Ct = xdbl[rowX + DTRANK + DSTATE + n];
        const float dA = __expf(dl * a);
        h = dA * h + (dl * ut) * Bt;
        float p = h * Ct;
        p += __shfl_xor(p, 1);
        p += __shfl_xor(p, 2);
        p += __shfl_xor(p, 4);
        p += __shfl_xor(p, 8);
        if (n == 0) {
            const float z = xz[rowZ];
            y[rowD] = (p + ut * Dd) * silu_f(z);
        }
        rowD += DINNER; rowX += XPROJ_N; rowZ += 2 * DINNER;
    }
}

// ---------------------------------------------------------------- launcher
extern "C" void kernel_launch(void* const* d_in, const int* in_sizes, int n_in,
                              void* d_out, int out_size, void* d_ws, size_t ws_size,
                              hipStream_t stream)
{
    (void)in_sizes; (void)n_in; (void)out_size; (void)ws_size;
    const float* x       = (const float*)d_in[0];
    const float* ln_g    = (const float*)d_in[1];
    const float* ln_b    = (const float*)d_in[2];
    const float* W_in    = (const float*)d_in[3];
    const float* conv_w  = (const float*)d_in[4];
    const float* conv_b  = (const float*)d_in[5];
    const float* W_xproj = (const float*)d_in[6];
    const float* W_dt    = (const float*)d_in[7];
    const float* b_dt    = (const float*)d_in[8];
    const float* A_log   = (const float*)d_in[9];
    const float* D_skip  = (const float*)d_in[10];
    const float* W_out   = (const float*)d_in[11];

    const size_t NT = (size_t)B_SZ * L_SZ;   // 4096 tokens
    float* ws    = (float*)d_ws;
    float* xn    = ws;                         // NT * DMODEL
    float* xz    = xn    + NT * DMODEL;        // NT * 2*DINNER
    float* uconv = xz    + NT * (2 * DINNER);  // NT * DINNER
    float* xdbl  = uconv + NT * DINNER;        // NT * XPROJ_N
    float* delta = xdbl  + NT * XPROJ_N;       // NT * DINNER
    float* yv    = delta + NT * DINNER;        // NT * DINNER

    // 1) LayerNorm
    ln_kernel<<<(int)NT, 256, 0, stream>>>(x, ln_g, ln_b, xn);

    // 2) xz = xn @ W_in   (4096 x 4096 x 1024)
    dim3 g1((2 * DINNER) / BN, (int)(NT / BM));
    gemm_wmma_f32<<<g1, 256, 0, stream>>>(xn, W_in, xz,
        (int)NT, 2 * DINNER, DMODEL, DMODEL, 2 * DINNER, 2 * DINNER,
        0, nullptr, nullptr, 0);

    // 3) u = SiLU(causal_dwconv(u_half(xz)) + conv_b)
    conv_silu_kernel<<<(int)((NT * DINNER) / 256), 256, 0, stream>>>(xz, conv_w, conv_b, uconv);

    // 4) x_dbl = u @ W_xproj   (4096 x 96 x 2048), partial-N path
    dim3 g2((XPROJ_N + BN - 1) / BN, (int)(NT / BM));
    gemm_wmma_f32<<<g2, 256, 0, stream>>>(uconv, W_xproj, xdbl,
        (int)NT, XPROJ_N, DINNER, DINNER, XPROJ_N, XPROJ_N,
        0, nullptr, nullptr, 0);

    // 5) delta = softplus(dt @ W_dt + b_dt)   (4096 x 2048 x 64), dt strided in x_dbl
    dim3 g3(DINNER / BN, (int)(NT / BM));
    gemm_wmma_f32<<<g3, 256, 0, stream>>>(xdbl, W_dt, delta,
        (int)NT, DINNER, DTRANK, XPROJ_N, DINNER, DINNER,
        1, b_dt, nullptr, 0);

    // 6) selective scan + D-skip + SiLU(z) gate
    scan_kernel<<<B_SZ * (DINNER / 16), 256, 0, stream>>>(
        uconv, delta, xdbl, A_log, D_skip, xz, yv);

    // 7) out = x + y @ W_out   (4096 x 1024 x 2048), residual fused
    dim3 g4(DMODEL / BN, (int)(NT / BM));
    gemm_wmma_f32<<<g4, 256, 0, stream>>>(yv, W_out, (float*)d_out,
        (int)NT, DMODEL, DINNER, DINNER, DMODEL, DMODEL,
        2, nullptr, x, DMODEL);
}